// Test_25202868092980
// MI455X (gfx1250) — compile-verified
//
#include <hip/hip_runtime.h>

// Bilinear interpolation of 20M random queries against a 512x512 float table.
// Memory-bound: ~240 MB streaming traffic (floor ~10.3 us @ 23.3 TB/s HBM),
// 1 MB table lives in the 192 MB L2. Strategy:
//   - float4-vectorized streaming loads/stores with non-temporal hints
//     (keep the gather table hot in cache, let the stream pass through)
//   - per-query table gathers as two 4-byte-aligned float2 loads
//     (adjacent corners in a row are contiguous)
//   - wave32-friendly 256-thread blocks

#define NR 512
#define NZ 512

typedef float v4f  __attribute__((ext_vector_type(4)));
// 4-byte-aligned float2: corners (iz0, iz0+1) are contiguous but only
// dword-aligned. Compiler emits global_load_b64 if the target allows
// dword-aligned 64-bit access, else splits into two b32 loads.
typedef float v2fu __attribute__((ext_vector_type(2), aligned(4)));

__device__ __forceinline__ float bilerp_one(const float* __restrict__ tt,
                                            float r, float z) {
    // ir0 = clip(floor(r), 0, NR-2); H=1, origin=0 -> inv = 1
    int ir0 = (int)floorf(r);
    int iz0 = (int)floorf(z);
    ir0 = ir0 < 0 ? 0 : (ir0 > (NR - 2) ? (NR - 2) : ir0);
    iz0 = iz0 < 0 ? 0 : (iz0 > (NZ - 2) ? (NZ - 2) : iz0);

    float x1 = (float)ir0;
    float y1 = (float)iz0;
    float x2 = x1 + 1.0f;
    float y2 = y1 + 1.0f;

    int idx = ir0 * NZ + iz0;
    v2fu lo = *(const v2fu*)(tt + idx);       // Q11 = lo.x, Q12 = lo.y
    v2fu hi = *(const v2fu*)(tt + idx + NZ);  // Q21 = hi.x, Q22 = hi.y

    float wx2 = x2 - r;   // (x2 - r)
    float wx1 = r - x1;   // (r - x1)
    float wy2 = y2 - z;   // (y2 - z)
    float wy1 = z - y1;   // (z - y1)

    return lo.x * wx2 * wy2
         + hi.x * wx1 * wy2
         + lo.y * wx2 * wy1
         + hi.y * wx1 * wy1;
}

__global__ void __launch_bounds__(256)
bilerp_vec4_kernel(const float* __restrict__ r,
                   const float* __restrict__ z,
                   const float* __restrict__ tt,
                   float* __restrict__ out,
                   int nvec) {
    int i = blockIdx.x * blockDim.x + threadIdx.x;
    if (i >= nvec) return;

    // Streaming traffic: non-temporal so the 1 MB gather table stays cached.
    v4f rv = __builtin_nontemporal_load((const v4f*)r + i);
    v4f zv = __builtin_nontemporal_load((const v4f*)z + i);

    v4f t;
    t.x = bilerp_one(tt, rv.x, zv.x);
    t.y = bilerp_one(tt, rv.y, zv.y);
    t.z = bilerp_one(tt, rv.z, zv.z);
    t.w = bilerp_one(tt, rv.w, zv.w);

    __builtin_nontemporal_store(t, (v4f*)out + i);
}

__global__ void __launch_bounds__(256)
bilerp_tail_kernel(const float* __restrict__ r,
                   const float* __restrict__ z,
                   const float* __restrict__ tt,
                   float* __restrict__ out,
                   int start, int n) {
    int i = start + blockIdx.x * blockDim.x + threadIdx.x;
    if (i >= n) return;
    float rv = __builtin_nontemporal_load(r + i);
    float zv = __builtin_nontemporal_load(z + i);
    float t  = bilerp_one(tt, rv, zv);
    __builtin_nontemporal_store(t, out + i);
}

extern "C" void kernel_launch(void* const* d_in, const int* in_sizes, int n_in,
                              void* d_out, int out_size, void* d_ws, size_t ws_size,
                              hipStream_t stream) {
    const float* r  = (const float*)d_in[0];
    const float* z  = (const float*)d_in[1];
    const float* tt = (const float*)d_in[2];
    float* out      = (float*)d_out;

    const int n    = in_sizes[0];     // 20,000,000
    const int nvec = n >> 2;          // float4 groups
    const int rem  = n - (nvec << 2);

    if (nvec > 0) {
        const int threads = 256;
        const int blocks  = (nvec + threads - 1) / threads;
        bilerp_vec4_kernel<<<blocks, threads, 0, stream>>>(r, z, tt, out, nvec);
    }
    if (rem > 0) {
        bilerp_tail_kernel<<<1, 256, 0, stream>>>(r, z, tt, out, nvec << 2, n);
    }
}